// RoadAwareDetector_30288109371781
// MI455X (gfx1250) — compile-verified
//
#include <hip/hip_runtime.h>
#include <hip/hip_bf16.h>

#define B_      16
#define N_      193536
#define C_      9
#define K_      1024      /* padded pre-NMS top-K (valid = 1000) */
#define KV_     1000
#define MAXDET_ 100
#define SCORE_T 0.05f
#define IOU_T   0.5f

typedef float v2f __attribute__((ext_vector_type(2)));
typedef float v8f __attribute__((ext_vector_type(8)));

__device__ __forceinline__ float sigmoidf_(float x) {
  return 1.0f / (1.0f + __expf(-x));
}

// ---------------------------------------------------------------------------
// Stage 1: decode anchors+deltas -> boxes (B,N,4)  (pure streaming)
// ---------------------------------------------------------------------------
__global__ __launch_bounds__(256) void decode_kernel(
    const float* __restrict__ anchors, const float* __restrict__ deltas,
    float* __restrict__ boxes) {
  long long i = (long long)blockIdx.x * blockDim.x + threadIdx.x;
  if (i >= (long long)B_ * N_) return;
  int n = (int)(i % N_);
  const float4 a = ((const float4*)anchors)[n];
  const float4 d = ((const float4*)deltas)[i];
  float acx = 0.5f * (a.x + a.z), acy = 0.5f * (a.y + a.w);
  float aw = fmaxf(a.z - a.x, 1.0f), ah = fmaxf(a.w - a.y, 1.0f);
  float cx = d.x * aw + acx;
  float cy = d.y * ah + acy;
  float w = __expf(fminf(d.z, 4.0f)) * aw;
  float h = __expf(fminf(d.w, 4.0f)) * ah;
  float4 o;
  o.x = cx - 0.5f * w; o.y = cy - 0.5f * h;
  o.z = cx + 0.5f * w; o.w = cy + 0.5f * h;
  ((float4*)boxes)[i] = o;
}

// ---------------------------------------------------------------------------
// Stage 2: per (b,c) select top-1000 scores (sorted desc) via 11-bit radix
// histogram threshold + LDS bitonic sort of 1024.  One 1024-thread block per
// (b,c); scores computed on the fly from logits (no 110MB score buffer).
// ---------------------------------------------------------------------------
__global__ __launch_bounds__(1024) void topk_kernel(
    const float* __restrict__ cls, const float* __restrict__ obj,
    const float* __restrict__ boxes,
    float* __restrict__ topBoxes, float* __restrict__ topScores) {
  __shared__ unsigned hist[2048];
  __shared__ unsigned keyS[K_];
  __shared__ unsigned idxS[K_];
  __shared__ unsigned tBucket, cntAbove, cHi, cTie;

  const int tid = threadIdx.x;
  const int bc  = blockIdx.x;        // b*C_ + c
  const int b   = bc / C_, c = bc % C_;

  for (int i = tid; i < 2048; i += 1024) hist[i] = 0;
  keyS[tid] = 0; idxS[tid] = 0;
  if (tid == 0) { cHi = 0; cTie = 0; }
  __syncthreads();

  const float* clsB = cls + (size_t)b * N_ * C_ + c;
  const float* objB = obj + (size_t)b * N_;

  // pass 1: histogram of top-11 float bits (positive floats order as uints)
  for (int n = tid; n < N_; n += 1024) {
    float s = sigmoidf_(clsB[(size_t)n * C_]) * sigmoidf_(objB[n]);
    if (s > SCORE_T) atomicAdd(&hist[__float_as_uint(s) >> 21], 1u);
  }
  __syncthreads();

  if (tid == 0) {
    unsigned cum = 0, t = 0, above = 0;
    for (int i = 2047; i >= 0; --i) {
      unsigned h = hist[i];
      if (cum + h >= KV_ || i == 0) { t = (unsigned)i; above = cum; break; }
      cum += h;
    }
    tBucket = t; cntAbove = above;
  }
  __syncthreads();
  const unsigned T = tBucket, above = cntAbove;

  // pass 2: compact candidates into LDS slots
  for (int n = tid; n < N_; n += 1024) {
    float s = sigmoidf_(clsB[(size_t)n * C_]) * sigmoidf_(objB[n]);
    if (s > SCORE_T) {
      unsigned key = __float_as_uint(s);
      unsigned bkt = key >> 21;
      if (bkt > T) {
        unsigned p = atomicAdd(&cHi, 1u);
        if (p < KV_) { keyS[p] = key; idxS[p] = (unsigned)n; }
      } else if (bkt == T) {
        unsigned p = above + atomicAdd(&cTie, 1u);
        if (p < KV_) { keyS[p] = key; idxS[p] = (unsigned)n; }
      }
    }
  }
  __syncthreads();

  // bitonic sort 1024 (descending by key), 1 element / thread
  for (unsigned k = 2; k <= K_; k <<= 1) {
    for (unsigned j = k >> 1; j > 0; j >>= 1) {
      unsigned ixj = (unsigned)tid ^ j;
      if (ixj > (unsigned)tid) {
        bool asc = ((unsigned)tid & k) != 0;
        unsigned ka = keyS[tid], kb = keyS[ixj];
        if (asc ? (ka > kb) : (ka < kb)) {
          keyS[tid] = kb; keyS[ixj] = ka;
          unsigned ia = idxS[tid]; idxS[tid] = idxS[ixj]; idxS[ixj] = ia;
        }
      }
      __syncthreads();
    }
  }

  // emit sorted boxes+scores (pad rows -> zero box / zero score)
  {
    unsigned key = (tid < KV_) ? keyS[tid] : 0u;
    unsigned idx = (tid < KV_) ? idxS[tid] : 0u;
    float sc = key ? __uint_as_float(key) : 0.0f;
    float4 bx;
    if (key) bx = ((const float4*)boxes)[(size_t)b * N_ + idx];
    else     { bx.x = 0.f; bx.y = 0.f; bx.z = 0.f; bx.w = 0.f; }
    ((float4*)topBoxes)[(size_t)bc * K_ + tid] = bx;
    topScores[(size_t)bc * K_ + tid] = sc;
  }
}

// ---------------------------------------------------------------------------
// Stage 3: NMS per (b,c).  Whole 1024x1024 suppression bitmask in LDS
// (128KB mask + 16KB boxes fits CDNA5's 320KB WGP LDS).  Wave w owns col-strip
// w (32 cols) and walks the upper-triangular row-tiles 0..2w+1; the WMMA B
// operand (col areas) is loop-invariant and hoisted.  Two
// v_wmma_f32_16x16x4_f32 per 16x32 strip compute the outer-sum area_i+area_j
// (rank-2 matmul A=[area_i,1], B=[1;area_j]); VALU does the intersection;
// wave32 ballot assembles mask words with no atomics; the greedy scan is
// lane-parallel over 32 mask words with __shfl keep-bit broadcast.
// All value selects are on loaded values (not loads) to avoid exec-mask
// save/restore divergence around the LDS reads feeding the WMMA.
// ---------------------------------------------------------------------------
__global__ __launch_bounds__(1024) void nms_kernel(
    const float* __restrict__ topBoxes, const float* __restrict__ topScores,
    float* __restrict__ keptScores) {
  extern __shared__ unsigned char smemRaw[];
  float4*   boxS  = (float4*)smemRaw;                 // K_ * 16B  = 16KB
  float*    areaS = (float*)(boxS + K_);              // K_ * 4B   = 4KB
  unsigned* maskS = (unsigned*)(areaS + K_);          // K_*32*4B  = 128KB
  unsigned* keepS = maskS + K_ * 32;                  // 32 * 4B

  const int tid = threadIdx.x;
  const int bc  = blockIdx.x;

  // gfx1250 async gather: each lane DMAs one 16B box row into LDS (ASYNCcnt)
  {
    const float4* gsrc = ((const float4*)topBoxes) + (size_t)bc * K_ + tid;
    unsigned ldsAddr = (unsigned)(size_t)(void*)(boxS + tid);
    unsigned long long gaddr = (unsigned long long)(size_t)gsrc;
    asm volatile("global_load_async_to_lds_b128 %0, %1, off"
                 :: "v"(ldsAddr), "v"(gaddr) : "memory");
    asm volatile("s_wait_asynccnt 0" : : : "memory");
  }
  for (int i = tid; i < K_ * 32; i += 1024) maskS[i] = 0;
  __syncthreads();
  {
    float4 bx = boxS[tid];
    areaS[tid] = (bx.z - bx.x) * (bx.w - bx.y);
  }
  __syncthreads();

  const int   lane = tid & 31;
  const int   wave = tid >> 5;
  const bool  lo   = lane < 16;
  const int   l15  = lane & 15;
  const float loF  = lo ? 1.0f : 0.0f;     // lane-half select as a value

  // wave == col-strip; hoist B operands (col areas) out of the row loop
  const int strip   = wave;
  const int colBase = strip << 5;
  const float aj0 = areaS[colBase + l15];        // unconditional LDS loads
  const float aj1 = areaS[colBase + 16 + l15];
  v2f bb0; bb0.x = loF; bb0.y = lo ? aj0 : 0.0f; // B: row0 = 1, row1 = area_j
  v2f bb1; bb1.x = loF; bb1.y = lo ? aj1 : 0.0f;

  const int rowTileMax = (strip << 1) + 1;       // rowBase <= colBase + 31
  for (int rowTile = 0; rowTile <= rowTileMax; ++rowTile) {
    const int rowBase = rowTile << 4;

    // A (16x4): col0 = area_i, col1 = 1.0
    const float ai = areaS[rowBase + l15];
    v2f a; a.x = lo ? ai : 0.0f; a.y = loF;
    v8f zero8 = {};
    v8f d0 = __builtin_amdgcn_wmma_f32_16x16x4_f32(false, a, false, bb0,
                                                   (short)0, zero8, false, false);
    v8f d1 = __builtin_amdgcn_wmma_f32_16x16x4_f32(false, a, false, bb1,
                                                   (short)0, zero8, false, false);

    const int i0 = rowBase + (lo ? 0 : 8);
    for (int v = 0; v < 8; ++v) {
      const int i = i0 + v;
      const float4 bi = boxS[i];

      int j = colBase + l15;
      float4 bj = boxS[j];
      float inter = fmaxf(fminf(bi.z, bj.z) - fmaxf(bi.x, bj.x), 0.0f) *
                    fmaxf(fminf(bi.w, bj.w) - fmaxf(bi.y, bj.y), 0.0f);
      float uni = fmaxf(d0[v] - inter, 1e-6f);
      bool sup0 = (j > i) && (inter > IOU_T * uni);
      unsigned bal0 = __builtin_amdgcn_ballot_w32(sup0);

      j = colBase + 16 + l15;
      bj = boxS[j];
      inter = fmaxf(fminf(bi.z, bj.z) - fmaxf(bi.x, bj.x), 0.0f) *
              fmaxf(fminf(bi.w, bj.w) - fmaxf(bi.y, bj.y), 0.0f);
      uni = fmaxf(d1[v] - inter, 1e-6f);
      bool sup1 = (j > i) && (inter > IOU_T * uni);
      unsigned bal1 = __builtin_amdgcn_ballot_w32(sup1);

      unsigned w0 = (bal0 & 0xFFFFu) | ((bal1 & 0xFFFFu) << 16); // row rowBase+v
      unsigned w1 = (bal0 >> 16) | (bal1 & 0xFFFF0000u);         // row rowBase+v+8
      if (lane < 2)
        maskS[(rowBase + v + (lane << 3)) * 32 + strip] = lane ? w1 : w0;
    }
  }
  __syncthreads();

  // greedy suppression scan: lane w owns keep word w
  if (tid < 32) {
    unsigned kw = 0xFFFFFFFFu;
    for (int i = 0; i < K_; ++i) {
      unsigned owner = __shfl(kw, i >> 5, 32);
      if ((owner >> (i & 31)) & 1u) kw &= ~maskS[i * 32 + tid];
    }
    keepS[tid] = kw;
  }
  __syncthreads();

  {
    float sc = topScores[(size_t)bc * K_ + tid];
    bool keep = (keepS[tid >> 5] >> (tid & 31)) & 1u;
    keptScores[(size_t)bc * K_ + tid] = keep ? sc : 0.0f;
  }
}

// ---------------------------------------------------------------------------
// Stage 4: per-image top-100 over 9x1024 kept scores -> (100,6) detections
// ---------------------------------------------------------------------------
__global__ __launch_bounds__(1024) void final_kernel(
    const float* __restrict__ keptScores, const float* __restrict__ topBoxes,
    float* __restrict__ out) {
  __shared__ unsigned hist[2048];
  __shared__ unsigned keyS[256];
  __shared__ unsigned idxS[256];
  __shared__ unsigned tBucket, cntAbove, cHi, cTie;

  const int tid = threadIdx.x;
  const int b   = blockIdx.x;

  for (int i = tid; i < 2048; i += 1024) hist[i] = 0;
  if (tid < 256) { keyS[tid] = 0; idxS[tid] = 0; }
  if (tid == 0) { cHi = 0; cTie = 0; }
  __syncthreads();

  const float* sc = keptScores + (size_t)b * C_ * K_;
  for (int f = tid; f < C_ * K_; f += 1024) {
    float s = sc[f];
    if (s > 0.0f) atomicAdd(&hist[__float_as_uint(s) >> 21], 1u);
  }
  __syncthreads();

  if (tid == 0) {
    unsigned cum = 0, t = 0, above = 0;
    for (int i = 2047; i >= 0; --i) {
      unsigned h = hist[i];
      if (cum + h >= MAXDET_ || i == 0) { t = (unsigned)i; above = cum; break; }
      cum += h;
    }
    tBucket = t; cntAbove = above;
  }
  __syncthreads();
  const unsigned T = tBucket, above = cntAbove;

  for (int f = tid; f < C_ * K_; f += 1024) {
    float s = sc[f];
    if (s > 0.0f) {
      unsigned key = __float_as_uint(s);
      unsigned bkt = key >> 21;
      if (bkt > T) {
        unsigned p = atomicAdd(&cHi, 1u);
        if (p < 256u) { keyS[p] = key; idxS[p] = (unsigned)f; }
      } else if (bkt == T) {
        unsigned p = above + atomicAdd(&cTie, 1u);
        if (p < 256u) { keyS[p] = key; idxS[p] = (unsigned)f; }
      }
    }
  }
  __syncthreads();

  for (unsigned k = 2; k <= 256; k <<= 1) {
    for (unsigned j = k >> 1; j > 0; j >>= 1) {
      if (tid < 256) {
        unsigned ixj = (unsigned)tid ^ j;
        if (ixj > (unsigned)tid) {
          bool asc = ((unsigned)tid & k) != 0;
          unsigned ka = keyS[tid], kb = keyS[ixj];
          if (asc ? (ka > kb) : (ka < kb)) {
            keyS[tid] = kb; keyS[ixj] = ka;
            unsigned ia = idxS[tid]; idxS[tid] = idxS[ixj]; idxS[ixj] = ia;
          }
        }
      }
      __syncthreads();
    }
  }

  if (tid < MAXDET_) {
    unsigned key = keyS[tid];
    unsigned f   = idxS[tid];
    int c = (int)(f >> 10);
    int r = (int)(f & (K_ - 1));
    float4 bx = ((const float4*)topBoxes)[((size_t)b * C_ + c) * K_ + r];
    float s = key ? __uint_as_float(key) : 0.0f;
    float* o = out + ((size_t)b * MAXDET_ + tid) * 6;
    o[0] = bx.x; o[1] = bx.y; o[2] = bx.z; o[3] = bx.w;
    o[4] = s;    o[5] = (float)c;
  }
}

// ---------------------------------------------------------------------------
extern "C" void kernel_launch(void* const* d_in, const int* in_sizes, int n_in,
                              void* d_out, int out_size, void* d_ws, size_t ws_size,
                              hipStream_t stream) {
  (void)in_sizes; (void)n_in; (void)out_size; (void)ws_size;
  const float* anchors = (const float*)d_in[0];
  const float* deltas  = (const float*)d_in[1];
  const float* cls     = (const float*)d_in[2];
  const float* obj     = (const float*)d_in[3];
  float* out = (float*)d_out;

  float* ws         = (float*)d_ws;
  float* boxes      = ws;                                   // B*N*4
  float* topBoxes   = boxes + (size_t)B_ * N_ * 4;          // B*C*K*4
  float* topScores  = topBoxes + (size_t)B_ * C_ * K_ * 4;  // B*C*K
  float* keptScores = topScores + (size_t)B_ * C_ * K_;     // B*C*K

  const size_t nmsLds = sizeof(float4) * K_ + sizeof(float) * K_ +
                        sizeof(unsigned) * (size_t)K_ * 32 + sizeof(unsigned) * 32;
  hipFuncSetAttribute((const void*)nms_kernel,
                      hipFuncAttributeMaxDynamicSharedMemorySize, (int)nmsLds);

  decode_kernel<<<dim3((B_ * N_ + 255) / 256), dim3(256), 0, stream>>>(
      anchors, deltas, boxes);
  topk_kernel<<<dim3(B_ * C_), dim3(1024), 0, stream>>>(
      cls, obj, boxes, topBoxes, topScores);
  nms_kernel<<<dim3(B_ * C_), dim3(1024), nmsLds, stream>>>(
      topBoxes, topScores, keptScores);
  final_kernel<<<dim3(B_), dim3(1024), 0, stream>>>(
      keptScores, topBoxes, out);
}